// SigCLoss_90683939488060
// MI455X (gfx1250) — compile-verified
//
#include <hip/hip_runtime.h>

typedef __attribute__((ext_vector_type(2))) float v2f;
typedef __attribute__((ext_vector_type(8))) float v8f;

#define TILE_M 128
#define TILE_N 128
#define KC 64
#define LDS_STRIDE 68   // 64 + 4 pad: 68*4=272B row stride -> 16B-aligned, conflict-free b64 reads

// ---- CDNA5 async global->LDS copy (no VGPR round trip, tracked by ASYNCcnt) ----
__device__ __forceinline__ void async_load_b128(const float* gptr, const float* lptr) {
    unsigned lds_off = (unsigned)(unsigned long long)(uintptr_t)lptr; // addr[31:0] == LDS offset
    unsigned long long ga = (unsigned long long)(uintptr_t)gptr;
    asm volatile("global_load_async_to_lds_b128 %0, %1, off"
                 :: "v"(lds_off), "v"(ga) : "memory");
}
__device__ __forceinline__ void wait_asynccnt0() {
    asm volatile("s_wait_asynccnt 0x0" ::: "memory");
}

// ---------------- init: zero the workspace accumulators ----------------
__global__ void sigclip_init_kernel(double* ws) {
    ws[0] = 0.0;                              // pos loss sum
    ws[1] = 0.0;                              // neg loss sum
    ((unsigned long long*)ws)[2] = 0ull;      // pos (match) count
}

// ---------------- main: fused GEMM + sigmoid loss + reduction ----------------
__global__ __launch_bounds__(256)
void sigclip_main_kernel(const float* __restrict__ A,   // first_features  [N][D]
                         const float* __restrict__ B,   // second_features [M][D]
                         const int*   __restrict__ la,  // first_label  [N]
                         const int*   __restrict__ lb,  // second_label [M]
                         const float* __restrict__ scale_p,
                         const float* __restrict__ bias_p,
                         double* __restrict__ ws,
                         int D) {
    __shared__ float As[2][TILE_M * LDS_STRIDE];
    __shared__ float Bs[2][TILE_N * LDS_STRIDE];
    __shared__ double sPos, sNeg;
    __shared__ unsigned long long sCnt;

    const int tid  = threadIdx.x;
    const int wave = tid >> 5;
    const int lane = tid & 31;
    const int lo   = lane & 15;
    const int hi   = lane >> 4;
    const int tileM = blockIdx.y * TILE_M;
    const int tileN = blockIdx.x * TILE_N;

    v8f acc[8];
    const v8f vzero = {0.f,0.f,0.f,0.f,0.f,0.f,0.f,0.f};
#pragma unroll
    for (int i = 0; i < 8; ++i) acc[i] = vzero;

    const int NCHUNK = D / KC;   // 4

    // prologue: async-stage chunk 0 into buffer 0 (16 b128 loads per thread)
#pragma unroll
    for (int i = 0; i < 8; ++i) {
        int idx = tid + i * 256;          // 0..2047
        int r   = idx >> 4;               // row 0..127
        int c4  = idx & 15;               // float4 column 0..15
        async_load_b128(A + (size_t)(tileM + r) * D + c4 * 4, &As[0][r * LDS_STRIDE + c4 * 4]);
        async_load_b128(B + (size_t)(tileN + r) * D + c4 * 4, &Bs[0][r * LDS_STRIDE + c4 * 4]);
    }

    for (int kc = 0; kc < NCHUNK; ++kc) {
        const int buf = kc & 1;

        // my async loads for this buffer have landed; barrier makes everyone's visible
        wait_asynccnt0();
        __syncthreads();

        // prefetch next chunk into the other buffer (safe: its readers passed the barrier)
        if (kc + 1 < NCHUNK) {
            const int nbuf = (kc + 1) & 1;
            const int kb   = (kc + 1) * KC;
#pragma unroll
            for (int i = 0; i < 8; ++i) {
                int idx = tid + i * 256;
                int r   = idx >> 4;
                int c4  = idx & 15;
                async_load_b128(A + (size_t)(tileM + r) * D + kb + c4 * 4,
                                &As[nbuf][r * LDS_STRIDE + c4 * 4]);
                async_load_b128(B + (size_t)(tileN + r) * D + kb + c4 * 4,
                                &Bs[nbuf][r * LDS_STRIDE + c4 * 4]);
            }
        }

        // fragment base pointers: A rows owned by this wave, B rows per 16-col subtile
        const float* arow = &As[buf][(wave * 16 + lo) * LDS_STRIDE + 2 * hi];
        const float* brow = &Bs[buf][lo * LDS_STRIDE + 2 * hi];

#pragma unroll 4
        for (int kk = 0; kk < KC; kk += 4) {
            v2f a = *(const v2f*)(arow + kk);
#pragma unroll
            for (int ns = 0; ns < 8; ++ns) {
                v2f b = *(const v2f*)(brow + ns * 16 * LDS_STRIDE + kk);
                acc[ns] = __builtin_amdgcn_wmma_f32_16x16x4_f32(
                    /*neg_a=*/false, a, /*neg_b=*/false, b,
                    /*c_mod=*/(short)0, acc[ns], /*reuse_a=*/false, /*reuse_b=*/false);
            }
        }
        __syncthreads();
    }

    // ---------------- fused loss epilogue ----------------
    const float scale = *scale_p;
    const float bias  = *bias_p;

    // C/D layout: VGPR r, lane -> element (m = r + 8*hi, n = lo) of the 16x16 tile
    int rlab[8];
#pragma unroll
    for (int r = 0; r < 8; ++r)
        rlab[r] = la[tileM + wave * 16 + 8 * hi + r];

    float pos_s = 0.f, neg_s = 0.f;
    int pos_c = 0;
#pragma unroll
    for (int ns = 0; ns < 8; ++ns) {
        int cl = lb[tileN + ns * 16 + lo];
#pragma unroll
        for (int r = 0; r < 8; ++r) {
            float logit = scale * acc[ns][r] + bias;
            bool match  = (rlab[r] == cl);
            // loss = -log sigmoid(label*logit) = softplus(-label*logit)
            float t    = match ? -logit : logit;
            float loss = fmaxf(t, 0.f) + __logf(1.f + __expf(-fabsf(t)));
            if (match) { pos_s += loss; pos_c += 1; }
            else       { neg_s += loss; }
        }
    }

    if (tid == 0) { sPos = 0.0; sNeg = 0.0; sCnt = 0ull; }
    __syncthreads();

    // wave32 shuffle reduction
#pragma unroll
    for (int off = 16; off > 0; off >>= 1) {
        pos_s += __shfl_down(pos_s, off);
        neg_s += __shfl_down(neg_s, off);
        pos_c += __shfl_down(pos_c, off);
    }
    if (lane == 0) {
        atomicAdd(&sPos, (double)pos_s);
        atomicAdd(&sNeg, (double)neg_s);
        atomicAdd(&sCnt, (unsigned long long)pos_c);
    }
    __syncthreads();
    if (tid == 0) {
        atomicAdd(&ws[0], sPos);
        atomicAdd(&ws[1], sNeg);
        atomicAdd((unsigned long long*)&ws[2], sCnt);
    }
}

// ---------------- finalize: normalize and emit scalar ----------------
__global__ void sigclip_final_kernel(const double* __restrict__ ws,
                                     float* __restrict__ out,
                                     long long total, int Nrows) {
    double pos = ws[0];
    double neg = ws[1];
    long long cnt = (long long)((const unsigned long long*)ws)[2];
    long long npos = cnt > 0 ? cnt : 1;
    long long nneg = (total - cnt) > 0 ? (total - cnt) : 1;
    double res = pos / (double)npos + (neg / (double)nneg) * (double)(Nrows - 1);
    out[0] = (float)res;
}

extern "C" void kernel_launch(void* const* d_in, const int* in_sizes, int n_in,
                              void* d_out, int out_size, void* d_ws, size_t ws_size,
                              hipStream_t stream) {
    const float* A  = (const float*)d_in[0];   // first_features  [N][D]
    const float* B  = (const float*)d_in[1];   // second_features [M][D]
    const int*   la = (const int*)d_in[2];     // first_label  [N]
    const int*   lb = (const int*)d_in[3];     // second_label [M]
    const float* sc = (const float*)d_in[4];   // logit_scale
    const float* bi = (const float*)d_in[5];   // logit_bias

    const int N = in_sizes[2];
    const int M = in_sizes[3];
    const int D = in_sizes[0] / N;

    double* ws = (double*)d_ws;

    sigclip_init_kernel<<<1, 1, 0, stream>>>(ws);

    dim3 grid(M / TILE_N, N / TILE_M);   // 64 x 64 blocks of 128x128 logits
    sigclip_main_kernel<<<grid, 256, 0, stream>>>(A, B, la, lb, sc, bi, ws, D);

    sigclip_final_kernel<<<1, 1, 0, stream>>>(ws, (float*)d_out, (long long)N * M, N);
}